// Mamba2_31516470018288
// MI455X (gfx1250) — compile-verified
//
#include <hip/hip_runtime.h>
#include <hip/hip_bf16.h>
#include <math.h>

typedef float v2f __attribute__((ext_vector_type(2)));
typedef float v8f __attribute__((ext_vector_type(8)));

#define D_MODEL   192
#define D_INNER   384
#define HEADDIM   12
#define NHEADS    32
#define DSTATE    16
#define CONV_DIM  416
#define D_IN_PROJ 832
#define BATCH     4
#define HH        128
#define WWID      128
#define LSEQ      (HH*WWID)      // 16384
#define MTOT      (BATCH*LSEQ)   // 65536
#define LN_EPS    1e-5f
#define KVCHUNK   128
#define NCHUNK    (LSEQ/KVCHUNK) // 128

__device__ __forceinline__ v8f wmma4(v2f a, v2f b, v8f c) {
  // D = A(16x4 f32) * B(4x16 f32) + C(16x16 f32), full fp32 precision
  return __builtin_amdgcn_wmma_f32_16x16x4_f32(false, a, false, b, (short)0, c,
                                               false, false);
}

// CDNA5 async memory->LDS copy (16B per lane), tracked by ASYNCcnt.
// Generic LDS pointer's low 32 bits are the LDS byte address (aperture scheme).
__device__ __forceinline__ void async_copy_b128(void* lds_dst, const void* gsrc) {
  unsigned loff = (unsigned)(unsigned long long)lds_dst;
  asm volatile("global_load_async_to_lds_b128 %0, %1, off"
               :: "v"(loff), "v"(gsrc) : "memory");
}
__device__ __forceinline__ void async_wait0() {
  asm volatile("s_wait_asynccnt 0x0" ::: "memory");
}

// ---------------------------------------------------------------------------
// Kernel 1: zxbcdt = u @ W_in^T   (M=65536, N=832, K=192)
// 8 waves/block, each wave -> one 16-row m-tile x 64 cols (4 WMMA n-tiles).
// Weight tile (64x192) staged to LDS via async-to-LDS (stride 196, no bank
// conflicts). Epilogue: [0,384)->z, [384,800)->xBC, [800,832)->softplus dt.
// ---------------------------------------------------------------------------
__global__ __launch_bounds__(256) void k_inproj_u(
    const float* __restrict__ U, const float* __restrict__ Win,
    const float* __restrict__ dt_bias,
    float* __restrict__ Z, float* __restrict__ XBC, float* __restrict__ DT)
{
  __shared__ __attribute__((aligned(16))) float lw[64 * 196];
  const int tid = threadIdx.x;
  const int n0 = blockIdx.y * 64;

  for (int i = tid; i < 64 * 48; i += 256) {   // 64 rows x 48 float4
    const int row = i / 48, kq = i % 48;
    async_copy_b128(&lw[row * 196 + kq * 4],
                    &Win[(size_t)(n0 + row) * D_MODEL + kq * 4]);
  }
  async_wait0();
  __syncthreads();

  const int wave = tid >> 5, lane = tid & 31;
  const int half = lane >> 4, l16 = lane & 15;
  const int m0 = (blockIdx.x * 8 + wave) * 16;
  const float* arow = U + (size_t)(m0 + l16) * D_MODEL;

  v8f acc[4] = {};
  for (int kk = 0; kk < D_MODEL; kk += 4) {
    const int k = kk + 2 * half;
    const v2f a = *(const v2f*)(arow + k);
#pragma unroll
    for (int j = 0; j < 4; ++j) {
      const v2f b = *(const v2f*)&lw[(j * 16 + l16) * 196 + k];
      acc[j] = wmma4(a, b, acc[j]);
    }
  }

#pragma unroll
  for (int j = 0; j < 4; ++j) {
    const int n = n0 + j * 16 + l16;
#pragma unroll
    for (int r = 0; r < 8; ++r) {
      const int row = m0 + r + 8 * half;
      const float v = acc[j][r];
      if (n < D_INNER) {
        Z[(size_t)row * D_INNER + n] = v;
      } else if (n < D_INNER + CONV_DIM) {
        XBC[(size_t)row * CONV_DIM + (n - D_INNER)] = v;
      } else {
        const int hidx = n - (D_INNER + CONV_DIM);
        const float xx = v + dt_bias[hidx];
        DT[(size_t)row * NHEADS + hidx] = (xx > 20.f) ? xx : log1pf(expf(xx));
      }
    }
  }
}

// ---------------------------------------------------------------------------
// Kernel 2: pre-conv Cq = t @ W_t_in[784:800]^T  (M=65536, N=16, K=192)
// ---------------------------------------------------------------------------
__global__ __launch_bounds__(256) void k_inproj_t(
    const float* __restrict__ T, const float* __restrict__ Wt,
    float* __restrict__ CQp)
{
  __shared__ __attribute__((aligned(16))) float lw[16 * 196];
  const int tid = threadIdx.x;
  for (int i = tid; i < 16 * 48; i += 256) {
    const int row = i / 48, kq = i % 48;
    async_copy_b128(&lw[row * 196 + kq * 4],
                    &Wt[(size_t)row * D_MODEL + kq * 4]);
  }
  async_wait0();
  __syncthreads();

  const int wave = tid >> 5, lane = tid & 31;
  const int half = lane >> 4, l16 = lane & 15;
  const int m0 = (blockIdx.x * 8 + wave) * 16;
  const float* arow = T + (size_t)(m0 + l16) * D_MODEL;

  v8f acc = {};
  for (int kk = 0; kk < D_MODEL; kk += 4) {
    const int k = kk + 2 * half;
    const v2f a = *(const v2f*)(arow + k);
    const v2f b = *(const v2f*)&lw[l16 * 196 + k];
    acc = wmma4(a, b, acc);
  }
  const int baser = m0 + 8 * half;
#pragma unroll
  for (int r = 0; r < 8; ++r)
    CQp[(size_t)(baser + r) * DSTATE + l16] = acc[r];
}

// ---------------------------------------------------------------------------
// Kernel 3: depthwise 3x3 SAME conv + SiLU over xBC; split into x / Bk.
// One block per spatial position; each thread handles 4 channels (float4
// taps -> one global_load_b128 per tap). 416 = 104 float4, split at 384.
// ---------------------------------------------------------------------------
__global__ __launch_bounds__(128) void k_conv_xbc(
    const float* __restrict__ XBCpre, const float* __restrict__ CW,
    const float* __restrict__ CB,
    float* __restrict__ X, float* __restrict__ BK)
{
  const int s = blockIdx.x;            // b*16384 + h*128 + w
  const int c4 = threadIdx.x;
  if (c4 >= CONV_DIM / 4) return;
  const int c = c4 * 4;
  const int b = s >> 14;
  const int hw = s & 16383;
  const int h = hw >> 7, w = hw & 127;

  float wq[4][9];
#pragma unroll
  for (int i = 0; i < 4; ++i)
#pragma unroll
    for (int t = 0; t < 9; ++t) wq[i][t] = CW[(c + i) * 9 + t];

  float a0 = CB[c], a1 = CB[c + 1], a2 = CB[c + 2], a3 = CB[c + 3];
#pragma unroll
  for (int dy = -1; dy <= 1; ++dy) {
    const int hh = h + dy;
    if (hh < 0 || hh >= HH) continue;
#pragma unroll
    for (int dx = -1; dx <= 1; ++dx) {
      const int ww = w + dx;
      if (ww < 0 || ww >= WWID) continue;
      const int t = (dy + 1) * 3 + (dx + 1);
      const float4 v = *(const float4*)
          &XBCpre[((size_t)(b * LSEQ + hh * WWID + ww)) * CONV_DIM + c];
      a0 = fmaf(v.x, wq[0][t], a0);
      a1 = fmaf(v.y, wq[1][t], a1);
      a2 = fmaf(v.z, wq[2][t], a2);
      a3 = fmaf(v.w, wq[3][t], a3);
    }
  }
  float4 o;
  o.x = a0 / (1.0f + expf(-a0));
  o.y = a1 / (1.0f + expf(-a1));
  o.z = a2 / (1.0f + expf(-a2));
  o.w = a3 / (1.0f + expf(-a3));
  if (c < D_INNER) *(float4*)&X[(size_t)s * D_INNER + c] = o;
  else             *(float4*)&BK[(size_t)s * DSTATE + (c - D_INNER)] = o;
}

// ---------------------------------------------------------------------------
// Kernel 4: depthwise conv + SiLU for the 16 Cq channels (orig ch 400..415),
// 4 channels per thread (float4 taps).
// ---------------------------------------------------------------------------
__global__ __launch_bounds__(256) void k_conv_cq(
    const float* __restrict__ CQpre, const float* __restrict__ CW,
    const float* __restrict__ CB, float* __restrict__ CQ)
{
  const int idx = blockIdx.x * 256 + threadIdx.x;   // MTOT*4 total
  if (idx >= MTOT * 4) return;
  const int s = idx >> 2, c = (idx & 3) * 4;
  const int b = s >> 14, hw = s & 16383, h = hw >> 7, w = hw & 127;
  const int cg = D_INNER + DSTATE + c;              // 400 + c

  float wq[4][9];
#pragma unroll
  for (int i = 0; i < 4; ++i)
#pragma unroll
    for (int t = 0; t < 9; ++t) wq[i][t] = CW[(cg + i) * 9 + t];

  float a0 = CB[cg], a1 = CB[cg + 1], a2 = CB[cg + 2], a3 = CB[cg + 3];
#pragma unroll
  for (int dy = -1; dy <= 1; ++dy) {
    const int hh = h + dy;
    if (hh < 0 || hh >= HH) continue;
#pragma unroll
    for (int dx = -1; dx <= 1; ++dx) {
      const int ww = w + dx;
      if (ww < 0 || ww >= WWID) continue;
      const int t = (dy + 1) * 3 + (dx + 1);
      const float4 v = *(const float4*)
          &CQpre[((size_t)(b * LSEQ + hh * WWID + ww)) * DSTATE + c];
      a0 = fmaf(v.x, wq[0][t], a0);
      a1 = fmaf(v.y, wq[1][t], a1);
      a2 = fmaf(v.z, wq[2][t], a2);
      a3 = fmaf(v.w, wq[3][t], a3);
    }
  }
  float4 o;
  o.x = a0 / (1.0f + expf(-a0));
  o.y = a1 / (1.0f + expf(-a1));
  o.z = a2 / (1.0f + expf(-a2));
  o.w = a3 / (1.0f + expf(-a3));
  *(float4*)&CQ[(size_t)s * DSTATE + c] = o;
}

// ---------------------------------------------------------------------------
// Kernel 5: partial KV per (b,h,chunk):  KVp[n,d] = sum_l Bk[l,n]*V[l,d]*dA[l]
// dA = dt_soft * exp(A_log)  (the two minus signs cancel in the reference).
// 192 threads = 16(n) x 12(d). Deterministic (no atomics).
// ---------------------------------------------------------------------------
__global__ __launch_bounds__(192) void k_kv(
    const float* __restrict__ BK, const float* __restrict__ X,
    const float* __restrict__ DT, const float* __restrict__ Alog,
    float* __restrict__ KVpart)
{
  __shared__ float sbk[KVCHUNK * DSTATE];
  __shared__ float svs[KVCHUNK * HEADDIM];
  const int blk = blockIdx.x;             // ((b*32+h)*128 + chunk)
  const int chunk = blk & (NCHUNK - 1);
  const int h = (blk >> 7) & (NHEADS - 1);
  const int b = blk >> 12;
  const int l0 = chunk * KVCHUNK;
  const int tid = threadIdx.x;
  const float eA = expf(Alog[h]);

  for (int i = tid; i < KVCHUNK * DSTATE; i += 192) {
    const int l = i >> 4, n = i & 15;
    sbk[i] = BK[((size_t)(b * LSEQ + l0 + l)) * DSTATE + n];
  }
  for (int i = tid; i < KVCHUNK * HEADDIM; i += 192) {
    const int l = i / HEADDIM, d = i % HEADDIM;
    const size_t row = (size_t)(b * LSEQ + l0 + l);
    svs[i] = X[row * D_INNER + h * HEADDIM + d] * (DT[row * NHEADS + h] * eA);
  }
  __syncthreads();

  const int n = tid / HEADDIM, d = tid % HEADDIM;
  float sum = 0.f;
  for (int l = 0; l < KVCHUNK; ++l)
    sum = fmaf(sbk[l * DSTATE + n], svs[l * HEADDIM + d], sum);
  KVpart[(size_t)blk * (DSTATE * HEADDIM) + tid] = sum;
}

// Fixed-order chunk reduction -> KV[b,h,n,d]
__global__ __launch_bounds__(256) void k_kvred(
    const float* __restrict__ KVpart, float* __restrict__ KV)
{
  const int idx = blockIdx.x * 256 + threadIdx.x;    // 24576 total
  if (idx >= BATCH * NHEADS * DSTATE * HEADDIM) return;
  const int bh = idx / (DSTATE * HEADDIM);
  const int nd = idx % (DSTATE * HEADDIM);
  float s = 0.f;
  for (int c = 0; c < NCHUNK; ++c)
    s += KVpart[((size_t)bh * NCHUNK + c) * (DSTATE * HEADDIM) + nd];
  KV[idx] = s;
}

// ---------------------------------------------------------------------------
// Kernel 6 (fused tail): y = Cq@KV + x*D  -> LayerNorm -> *z -> @W_out^T.
// KV/Cq tiles staged via async-to-LDS; y lives in LDS (stride 390:
// conflict-free WMMA A reads). Out GEMM: 12 n-tiles over 8 waves, K=384.
// ---------------------------------------------------------------------------
__global__ __launch_bounds__(256) void k_final(
    const float* __restrict__ X, const float* __restrict__ CQ,
    const float* __restrict__ KV, const float* __restrict__ Z,
    const float* __restrict__ Dp, const float* __restrict__ gamma,
    const float* __restrict__ beta, const float* __restrict__ Wout,
    float* __restrict__ OUT)
{
  __shared__ float ylds[16 * 390];
  __shared__ __attribute__((aligned(16))) float kvs[NHEADS * DSTATE * HEADDIM];
  __shared__ __attribute__((aligned(16))) float cqs[16 * DSTATE];
  __shared__ float red[16 * 16];
  __shared__ float mu[16];
  __shared__ float rsd[16];

  const int tid = threadIdx.x;
  const int m0 = blockIdx.x * 16;
  const int b = m0 / LSEQ;                  // all 16 rows share one batch

  for (int i = tid; i < (NHEADS * DSTATE * HEADDIM) / 4; i += 256)  // 1536
    async_copy_b128(&kvs[i * 4],
                    &KV[(size_t)b * NHEADS * DSTATE * HEADDIM + i * 4]);
  for (int i = tid; i < (16 * DSTATE) / 4; i += 256)                // 64
    async_copy_b128(&cqs[i * 4], &CQ[(size_t)m0 * DSTATE + i * 4]);
  async_wait0();
  __syncthreads();

  // y = Cq @ KV + V*D
  for (int i = tid; i < 16 * D_INNER; i += 256) {
    const int m = i / D_INNER, c = i % D_INNER;
    const int h = c / HEADDIM, d = c % HEADDIM;
    float y = X[(size_t)(m0 + m) * D_INNER + c] * Dp[h];
    const float* kvrow = &kvs[h * DSTATE * HEADDIM + d];
    const float* cqrow = &cqs[m * DSTATE];
#pragma unroll
    for (int n = 0; n < DSTATE; ++n)
      y = fmaf(cqrow[n], kvrow[n * HEADDIM], y);
    ylds[m * 390 + c] = y;
  }
  __syncthreads();

  // LayerNorm over 384 channels (16 partials per row)
  {
    const int r = tid >> 4, t16 = tid & 15;
    float s = 0.f;
    for (int c = t16; c < D_INNER; c += 16) s += ylds[r * 390 + c];
    red[r * 16 + t16] = s;
    __syncthreads();
    if (tid < 16) {
      float m = 0.f;
      for (int i = 0; i < 16; ++i) m += red[tid * 16 + i];
      mu[tid] = m * (1.0f / D_INNER);
    }
    __syncthreads();
    const float mur = mu[r];
    float v = 0.f;
    for (int c = t16; c < D_INNER; c += 16) {
      const float dd = ylds[r * 390 + c] - mur;
      v = fmaf(dd, dd, v);
    }
    red[r * 16 + t16] = v;
    __syncthreads();
    if (tid < 16) {
      float m = 0.f;
      for (int i = 0; i < 16; ++i) m += red[tid * 16 + i];
      rsd[tid] = rsqrtf(m * (1.0f / D_INNER) + LN_EPS);
    }
    __syncthreads();
  }
  for (int i = tid; i < 16 * D_INNER; i += 256) {
    const int m = i / D_INNER, c = i % D_INNER;
    float y = (ylds[m * 390 + c] - mu[m]) * rsd[m] * gamma[c] + beta[c];
    y *= Z[(size_t)(m0 + m) * D_INNER + c];
    ylds[m * 390 + c] = y;
  }
  __syncthreads();

  // out = yz @ W_out^T  (16 x 192, K=384) with fp32 WMMA
  const int wave = tid >> 5, lane = tid & 31;
  const int half = lane >> 4, l16 = lane & 15;
  for (int nt = wave; nt < 12; nt += 8) {
    const int n0 = nt * 16;
    const float* wrow = Wout + (size_t)(n0 + l16) * D_INNER;
    v8f acc = {};
    for (int kk = 0; kk < D_INNER; kk += 4) {
      const int k = kk + 2 * half;
      const v2f a = *(const v2f*)&ylds[l16 * 390 + k];
      const v2f bb = *(const v2f*)(wrow + k);
      acc = wmma4(a, bb, acc);
    }
    const int baser = m0 + 8 * half;
#pragma unroll
    for (int r = 0; r < 8; ++r)
      OUT[(size_t)(baser + r) * D_MODEL + n0 + l16] = acc[r];
  }
}

// ---------------------------------------------------------------------------
extern "C" void kernel_launch(void* const* d_in, const int* in_sizes, int n_in,
                              void* d_out, int out_size, void* d_ws, size_t ws_size,
                              hipStream_t stream)
{
  (void)in_sizes; (void)n_in; (void)out_size; (void)ws_size;
  const float* u      = (const float*)d_in[0];
  const float* t      = (const float*)d_in[1];
  const float* Win    = (const float*)d_in[2];
  const float* Wtin   = (const float*)d_in[3];
  const float* convw  = (const float*)d_in[4];
  const float* convb  = (const float*)d_in[5];
  const float* dtbias = (const float*)d_in[6];
  const float* Alog   = (const float*)d_in[7];
  const float* Dp     = (const float*)d_in[8];
  const float* gamma  = (const float*)d_in[9];
  const float* beta   = (const float*)d_in[10];
  const float* Wout   = (const float*)d_in[11];
  float* out = (float*)d_out;

  float* ws = (float*)d_ws;
  size_t o = 0;
  float* Z    = ws + o; o += (size_t)MTOT * D_INNER;     // 65536*384
  float* XBCp = ws + o; o += (size_t)MTOT * CONV_DIM;    // 65536*416
  float* DT   = ws + o; o += (size_t)MTOT * NHEADS;      // 65536*32 (softplus'd)
  float* CQp  = ws + o; o += (size_t)MTOT * DSTATE;      // pre-conv Cq
  float* Xp   = ws + o; o += (size_t)MTOT * D_INNER;     // post-conv x (=V)
  float* BK   = ws + o; o += (size_t)MTOT * DSTATE;      // post-conv B
  float* CQ   = ws + o; o += (size_t)MTOT * DSTATE;      // post-conv C
  float* KVp  = ws + o; o += (size_t)BATCH * NHEADS * NCHUNK * DSTATE * HEADDIM;
  float* KV   = ws + o; o += (size_t)BATCH * NHEADS * DSTATE * HEADDIM;

  const dim3 b256(256);
  k_inproj_u<<<dim3(MTOT / 128, D_IN_PROJ / 64), b256, 0, stream>>>(
      u, Win, dtbias, Z, XBCp, DT);
  // Only W_t_in rows 784..799 matter (post-slice channels 400..415 = Cq)
  k_inproj_t<<<dim3(MTOT / 128), b256, 0, stream>>>(
      t, Wtin + (size_t)(2 * D_INNER + DSTATE) * D_MODEL, CQp);
  k_conv_xbc<<<dim3(MTOT), dim3(128), 0, stream>>>(XBCp, convw, convb, Xp, BK);
  k_conv_cq<<<dim3((MTOT * 4) / 256), b256, 0, stream>>>(CQp, convw, convb, CQ);
  k_kv<<<dim3(BATCH * NHEADS * NCHUNK), dim3(192), 0, stream>>>(BK, Xp, DT, Alog, KVp);
  k_kvred<<<dim3((BATCH * NHEADS * DSTATE * HEADDIM + 255) / 256), b256, 0, stream>>>(KVp, KV);
  k_final<<<dim3(MTOT / 16), b256, 0, stream>>>(
      Xp, CQ, KV, Z, Dp, gamma, beta, Wout, out);
}